// LinkPredictor_87900800680116
// MI455X (gfx1250) — compile-verified
//
#include <hip/hip_runtime.h>
#include <hip/hip_bf16.h>

typedef __attribute__((ext_vector_type(2))) float v2f;
typedef __attribute__((ext_vector_type(8))) float v8f;

// ======================================================================
// GEMM: Out[M x 256] = X[M x K] @ [ Wl | Wr ]   (Wl, Wr each K x 128, row-major)
// One wave32 computes one 16x16 output tile with V_WMMA_F32_16X16X4_F32.
// A-frag layout: lane L, vgpr j -> M = L&15,  K = k + j + 2*(L>>4)  (contiguous float2)
// B-frag layout: lane L, vgpr j -> N = L&15,  K = k + j + 2*(L>>4)
// C/D layout:    lane L, vgpr j -> N = L&15,  M = j + 8*(L>>4)
// ======================================================================
__global__ void gemm_dual_wmma(const float* __restrict__ X,
                               const float* __restrict__ Wl,
                               const float* __restrict__ Wr,
                               float* __restrict__ Out,
                               int M, int K)
{
    const int wave = threadIdx.x >> 5;
    const int lane = threadIdx.x & 31;
    const int tile = blockIdx.x * (blockDim.x >> 5) + wave;
    const int mt = tile >> 4;            // 16 column tiles (256 cols)
    const int nt = tile & 15;
    const int m0 = mt * 16;
    if (m0 >= M) return;                 // wave-uniform, EXEC stays all-ones
    const int n0 = nt * 16;

    const int lo = lane & 15;
    const int hi = lane >> 4;

    const float* __restrict__ Wp = (n0 < 128) ? Wl : Wr;
    const int bcol = (n0 & 127) + lo;

    const float* arow = X + (size_t)(m0 + lo) * K + 2 * hi;      // 8B-aligned
    const float* bptr = Wp + (size_t)(2 * hi) * 128 + bcol;

    v8f c = {0.f, 0.f, 0.f, 0.f, 0.f, 0.f, 0.f, 0.f};
    #pragma unroll 4
    for (int k = 0; k < K; k += 4) {
        v2f a = *(const v2f*)arow;                 // K = k+2*hi, k+2*hi+1
        v2f b;
        b.x = bptr[0];                             // K = k+2*hi
        b.y = bptr[128];                           // K = k+2*hi+1
        c = __builtin_amdgcn_wmma_f32_16x16x4_f32(
                /*neg_a=*/false, a, /*neg_b=*/false, b,
                /*c_mod=*/(short)0, c, /*reuse_a=*/false, /*reuse_b=*/false);
        arow += 4;
        bptr += 4 * 128;
    }

    float* orow = Out + (size_t)(m0 + 8 * hi) * 256 + n0 + lo;
    #pragma unroll
    for (int j = 0; j < 8; ++j)
        orow[(size_t)j * 256] = c[j];
}

// ======================================================================
// Edge scatter: agg[dst] += Y[src, 0:128]  (Y has row stride 256),
// optionally cnt[dst] += 1.  One thread handles 4 channels (float4).
// ======================================================================
__global__ void scatter_add_kernel(const float* __restrict__ Y,
                                   const int* __restrict__ src,
                                   const int* __restrict__ dst,
                                   float* __restrict__ agg,
                                   float* __restrict__ cnt,
                                   int E, int add_count)
{
    long tid = (long)blockIdx.x * blockDim.x + threadIdx.x;
    int e = (int)(tid >> 5);
    if (e >= E) return;
    int c4 = ((int)tid & 31) << 2;
    int s = src[e];
    int d = dst[e];
    const float4 v = *(const float4*)(Y + (size_t)s * 256 + c4);
    float* a = agg + (size_t)d * 128 + c4;
    unsafeAtomicAdd(a + 0, v.x);
    unsafeAtomicAdd(a + 1, v.y);
    unsafeAtomicAdd(a + 2, v.z);
    unsafeAtomicAdd(a + 3, v.w);
    if (add_count && c4 == 0) unsafeAtomicAdd(cnt + d, 1.0f);
}

// ======================================================================
// H[i,c] = (agg[i,c]/max(cnt[i],1) + bias[c] + Y[i, 128+c]) , optional relu
// ======================================================================
__global__ void finalize_kernel(const float* __restrict__ agg,
                                const float* __restrict__ cnt,
                                const float* __restrict__ bias,
                                const float* __restrict__ Ymat,
                                float* __restrict__ H,
                                int N, int do_relu)
{
    long tid = (long)blockIdx.x * blockDim.x + threadIdx.x;
    if (tid >= (long)N * 128) return;
    int i = (int)(tid >> 7);
    int c = (int)tid & 127;
    float inv = 1.0f / fmaxf(cnt[i], 1.0f);
    float v = agg[tid] * inv + bias[c] + Ymat[(size_t)i * 256 + 128 + c];
    if (do_relu) v = fmaxf(v, 0.0f);
    H[tid] = v;
}

// ======================================================================
// Per-node link scores: sab[i,0] = h2[i,:] . Wlin[0:128]
//                       sab[i,1] = h2[i,:] . Wlin[128:256]
// One wave per node; wave32 shuffle reduction.
// ======================================================================
__global__ void node_scores_kernel(const float* __restrict__ H,
                                   const float* __restrict__ Wlin,
                                   float* __restrict__ sab, int N)
{
    int node = blockIdx.x * (blockDim.x >> 5) + (threadIdx.x >> 5);
    int lane = threadIdx.x & 31;
    if (node >= N) return;
    const float* row = H + (size_t)node * 128;
    float a = 0.f, b = 0.f;
    #pragma unroll
    for (int c = lane; c < 128; c += 32) {
        float hv = row[c];
        a += hv * Wlin[c];
        b += hv * Wlin[128 + c];
    }
    #pragma unroll
    for (int off = 16; off > 0; off >>= 1) {
        a += __shfl_down(a, off, 32);
        b += __shfl_down(b, off, 32);
    }
    if (lane == 0) {
        sab[(size_t)node * 2 + 0] = a;
        sab[(size_t)node * 2 + 1] = b;
    }
}

// ======================================================================
// out[e] = sab[e0[e],0] + sab[e1[e],1] + b_lin
// ======================================================================
__global__ void edge_scores_kernel(const float* __restrict__ sab,
                                   const int* __restrict__ e0,
                                   const int* __restrict__ e1,
                                   const float* __restrict__ b_lin,
                                   float* __restrict__ out, int E)
{
    int i = blockIdx.x * blockDim.x + threadIdx.x;
    if (i >= E) return;
    out[i] = sab[(size_t)e0[i] * 2] + sab[(size_t)e1[i] * 2 + 1] + b_lin[0];
}

extern "C" void kernel_launch(void* const* d_in, const int* in_sizes, int n_in,
                              void* d_out, int out_size, void* d_ws, size_t ws_size,
                              hipStream_t stream)
{
    const float* x     = (const float*)d_in[0];
    const int*   eidx  = (const int*)d_in[1];
    const int*   pos   = (const int*)d_in[2];
    const int*   neg   = (const int*)d_in[3];
    const float* W_l1  = (const float*)d_in[4];
    const float* b_l1  = (const float*)d_in[5];
    const float* W_r1  = (const float*)d_in[6];
    const float* W_l2  = (const float*)d_in[7];
    const float* b_l2  = (const float*)d_in[8];
    const float* W_r2  = (const float*)d_in[9];
    const float* W_lin = (const float*)d_in[10];
    const float* b_lin = (const float*)d_in[11];

    const int N  = 100000;
    const int E  = in_sizes[1] / 2;
    const int Ep = in_sizes[2] / 2;
    const int En = in_sizes[3] / 2;

    float* buf_y   = (float*)d_ws;                       // N*256
    float* buf_agg = buf_y   + (size_t)N * 256;          // N*128
    float* buf_h   = buf_agg + (size_t)N * 128;          // N*128
    float* cnt     = buf_h   + (size_t)N * 128;          // N
    float* sab     = cnt     + N;                        // N*2
    float* outp    = (float*)d_out;

    const int tiles   = (N / 16) * 16;                   // 16x16 tiles over N x 256
    const int gblocks = tiles / 4;                       // 4 waves / block
    const int sblocks = (int)(((long)E * 32 + 255) / 256);
    const int fblocks = (int)(((long)N * 128 + 255) / 256);

    // ---------------- layer 1 ----------------
    hipMemsetAsync(buf_agg, 0, (size_t)N * 128 * sizeof(float), stream);
    hipMemsetAsync(cnt,     0, (size_t)N * sizeof(float), stream);

    gemm_dual_wmma<<<gblocks, 128, 0, stream>>>(x, W_l1, W_r1, buf_y, N, 384);
    scatter_add_kernel<<<sblocks, 256, 0, stream>>>(buf_y, eidx, eidx + E,
                                                    buf_agg, cnt, E, 1);
    finalize_kernel<<<fblocks, 256, 0, stream>>>(buf_agg, cnt, b_l1, buf_y,
                                                 buf_h, N, 1);

    // ---------------- layer 2 ----------------
    gemm_dual_wmma<<<gblocks, 128, 0, stream>>>(buf_h, W_l2, W_r2, buf_y, N, 128);
    hipMemsetAsync(buf_agg, 0, (size_t)N * 128 * sizeof(float), stream);
    scatter_add_kernel<<<sblocks, 256, 0, stream>>>(buf_y, eidx, eidx + E,
                                                    buf_agg, cnt, E, 0);
    finalize_kernel<<<fblocks, 256, 0, stream>>>(buf_agg, cnt, b_l2, buf_y,
                                                 buf_h, N, 0);

    // ---------------- link scoring ----------------
    node_scores_kernel<<<(N + 7) / 8, 256, 0, stream>>>(buf_h, W_lin, sab, N);
    edge_scores_kernel<<<(Ep + 255) / 256, 256, 0, stream>>>(sab, pos, pos + Ep,
                                                             b_lin, outp, Ep);
    edge_scores_kernel<<<(En + 255) / 256, 256, 0, stream>>>(sab, neg, neg + En,
                                                             b_lin, outp + Ep, En);
}